// BaseHeads_69965017252515
// MI455X (gfx1250) — compile-verified
//
#include <hip/hip_runtime.h>

// Problem constants (match reference): B=2, S=128, H=768, R=4
#define B_   2
#define S_   128
#define H_   768
#define R_   4
#define NOUT (H_ * R_)   // 3072 projected channels
#define M_   (B_ * S_)   // 256 rows of x

typedef __attribute__((ext_vector_type(2))) float v2f;
typedef __attribute__((ext_vector_type(8))) float v8f;

// ---------------------------------------------------------------------------
// Fast native tanh: CDNA5 has V_TANH_F32 (TRANS unit, co-executes with VALU).
// ---------------------------------------------------------------------------
__device__ __forceinline__ float fast_tanh(float x) {
#if __has_builtin(__builtin_amdgcn_tanhf)
  return __builtin_amdgcn_tanhf(x);
#else
  float r;
  asm("v_tanh_f32 %0, %1" : "=v"(r) : "v"(x));
  return r;
#endif
}

// ---------------------------------------------------------------------------
// Phase 1: hs = x * w_src^T + b_src ; ht = x * w_tgt^T + b_tgt
// Full-fp32 GEMM via V_WMMA_F32_16X16X4_F32 (exact f32, matches reference).
// One wave per 16x16 output tile. K = 768, stepped by 4.
//
// Per ISA 7.12.2 layouts (32-bit):
//   A 16x4 : lane holds row m=lane%16, k = 2*(lane/16) + {0,1}  (contig pair)
//   B 4x16 : lane holds col n=lane%16, k = 2*(lane/16) + {0,1}  (contig pair)
//   D 16x16: VGPR r -> row (r + 8*(lane/16)), col n=lane%16
// ---------------------------------------------------------------------------
__global__ __launch_bounds__(256) void proj_wmma_f32(
    const float* __restrict__ x,
    const float* __restrict__ w_src, const float* __restrict__ b_src,
    const float* __restrict__ w_tgt, const float* __restrict__ b_tgt,
    float* __restrict__ hs, float* __restrict__ ht) {
  const int wave  = threadIdx.x >> 5;
  const int lane  = threadIdx.x & 31;
  const int ntile = blockIdx.x * 8 + wave;   // 0..191  (3072/16)
  const int mtile = blockIdx.y;              // 0..15   (256/16)

  const float* w;
  const float* bias;
  float* outp;
  if (blockIdx.z == 0) { w = w_src; bias = b_src; outp = hs; }
  else                 { w = w_tgt; bias = b_tgt; outp = ht; }

  const int lm    = lane & 15;
  const int khalf = (lane >> 4) * 2;
  const float* arow = x + (size_t)(mtile * 16 + lm) * H_ + khalf;  // row of x
  const float* brow = w + (size_t)(ntile * 16 + lm) * H_ + khalf;  // row of W

  v8f acc = {0.f, 0.f, 0.f, 0.f, 0.f, 0.f, 0.f, 0.f};
  for (int k = 0; k < H_; k += 4) {
    v2f a = *(const v2f*)(arow + k);
    v2f b = *(const v2f*)(brow + k);
    acc = __builtin_amdgcn_wmma_f32_16x16x4_f32(
        /*neg_a=*/false, a, /*neg_b=*/false, b,
        /*c_mod=*/(short)0, acc, /*reuse_a=*/false, /*reuse_b=*/false);
  }

  const int   n     = ntile * 16 + lm;
  const float bv    = bias[n];
  const int   rbase = mtile * 16 + (lane >> 4) * 8;
#pragma unroll
  for (int r = 0; r < 8; ++r) {
    outp[(size_t)(rbase + r) * NOUT + n] = acc[r] + bv;
  }
}

// ---------------------------------------------------------------------------
// Phase 2: out[b,r,i,j] = sum_h tanh(hs[b,i,r,h] + ht[b,j,r,h]) * w_out[h]
// 256-thread block owns a 16x16 (i,j) tile at fixed (b,r). Rows staged in LDS
// (row stride padded to 772 floats -> j-stride hits distinct bank groups).
// Inner loop: float4 LDS loads, 4x v_tanh_f32 (TRANS) + 4x FMA (VALU).
// ---------------------------------------------------------------------------
#define TI 16
#define TJ 16
#define RS 772  // padded LDS row stride (floats); 772*4 B is 16B-aligned

__global__ __launch_bounds__(256) void pair_tanh_reduce(
    const float* __restrict__ hs, const float* __restrict__ ht,
    const float* __restrict__ w_out, float* __restrict__ out) {
  __shared__ float s_a[TI * RS];
  __shared__ float s_b[TJ * RS];
  __shared__ float s_w[H_];

  const int br = blockIdx.x;       // b*R + r
  const int b  = br >> 2;
  const int r  = br & 3;
  const int i0 = blockIdx.y * TI;
  const int j0 = blockIdx.z * TJ;
  const int t  = threadIdx.x;

  // Stage tiles: 16 rows x 768 floats each = 3072 float4 per matrix.
  for (int idx = t; idx < TI * (H_ / 4); idx += 256) {
    const int row  = idx / (H_ / 4);
    const int col4 = (idx % (H_ / 4)) * 4;
    const float4 va =
        *(const float4*)(hs + (size_t)(b * S_ + i0 + row) * NOUT + r * H_ + col4);
    *(float4*)&s_a[row * RS + col4] = va;
    const float4 vb =
        *(const float4*)(ht + (size_t)(b * S_ + j0 + row) * NOUT + r * H_ + col4);
    *(float4*)&s_b[row * RS + col4] = vb;
  }
  for (int idx = t; idx < H_ / 4; idx += 256) {
    *(float4*)&s_w[idx * 4] = *(const float4*)(w_out + idx * 4);
  }
  __syncthreads();

  const int i = t >> 4;
  const int j = t & 15;
  const float* pa = &s_a[i * RS];
  const float* pb = &s_b[j * RS];

  float acc = 0.f;
  for (int h = 0; h < H_; h += 4) {
    const float4 a = *(const float4*)(pa + h);
    const float4 c = *(const float4*)(pb + h);
    const float4 w = *(const float4*)(&s_w[h]);
    acc += fast_tanh(a.x + c.x) * w.x;
    acc += fast_tanh(a.y + c.y) * w.y;
    acc += fast_tanh(a.z + c.z) * w.z;
    acc += fast_tanh(a.w + c.w) * w.w;
  }

  out[((size_t)br * S_ + (i0 + i)) * S_ + (j0 + j)] = acc;
}

// ---------------------------------------------------------------------------
// Launch
// ---------------------------------------------------------------------------
extern "C" void kernel_launch(void* const* d_in, const int* in_sizes, int n_in,
                              void* d_out, int out_size, void* d_ws, size_t ws_size,
                              hipStream_t stream) {
  (void)in_sizes; (void)n_in; (void)out_size; (void)ws_size;
  const float* x     = (const float*)d_in[0];
  const float* w_src = (const float*)d_in[1];
  const float* b_src = (const float*)d_in[2];
  const float* w_tgt = (const float*)d_in[3];
  const float* b_tgt = (const float*)d_in[4];
  const float* w_out = (const float*)d_in[5];
  float* out = (float*)d_out;

  float* hs = (float*)d_ws;                       // (256, 3072) = 3 MB
  float* ht = hs + (size_t)M_ * NOUT;             // (256, 3072) = 3 MB

  // Phase 1: 192 ntiles (8 waves/block -> 24 blocks) x 16 mtiles x 2 matrices.
  proj_wmma_f32<<<dim3(24, 16, 2), 256, 0, stream>>>(
      x, w_src, b_src, w_tgt, b_tgt, hs, ht);

  // Phase 2: (B*R=8) x (S/16=8) x (S/16=8) blocks of 256 threads.
  pair_tanh_reduce<<<dim3(B_ * R_, S_ / TI, S_ / TJ), 256, 0, stream>>>(
      hs, ht, w_out, out);
}